// YOLOv3Loss_63780264346014
// MI455X (gfx1250) — compile-verified
//
#include <hip/hip_runtime.h>
#include <math.h>

// ---------------------------------------------------------------------------
// YOLOv3 loss, MI455X (gfx1250, wave32).
//  - zero kernel: clears cell->target maps in d_ws and d_out
//  - scatter kernel (per scale): 64x50 targets -> best-anchor cell map
//  - main kernel (per scale): one thread per anchor cell; no-obj fast path
//    reads only the conf logit; obj cells do full coord/conf/cls math.
//    Wave reduction uses V_WMMA_F32_16X16X4_F32 (exact f32).
// ---------------------------------------------------------------------------

typedef float v2f __attribute__((ext_vector_type(2)));
typedef float v8f __attribute__((ext_vector_type(8)));

#define BCE_EPS 1e-12f
#define NCLS 20
#define BATCH 64
#define NTGT 50

__device__ __forceinline__ float sigm(float x) {
  return 1.0f / (1.0f + expf(-x));
}

__device__ __forceinline__ float safelog(float p) {
  // log(clip(p, eps, 1.0))
  return logf(fminf(fmaxf(p, BCE_EPS), 1.0f));
}

// Exact f32 sum of all 32 lanes' `val` using WMMA f32 16x16x4.
// A layout (32-bit A 16x4): VGPR0 lanes0-15 -> A[m][0], lanes16-31 -> A[m][2].
// With A[.][0]=x[lane], A[.][2]=x[lane+16], A[.][1]=A[.][3]=0 and B = ones,
// D[m][n] = x[m] + x[m+16]. D VGPR j: lanes0-15 hold D[j][n], lanes16-31 hold
// D[j+8][n]. Sum of the 8 D VGPRs + cross-half shuffle => full wave sum.
__device__ __forceinline__ float wave_sum_wmma(float val) {
  v2f av; av[0] = val;  av[1] = 0.0f;
  v2f bv; bv[0] = 1.0f; bv[1] = 1.0f;
  v8f cv = {};
  cv = __builtin_amdgcn_wmma_f32_16x16x4_f32(
      /*neg_a=*/false, av, /*neg_b=*/false, bv,
      /*c_mod=*/(short)0, cv, /*reuse_a=*/false, /*reuse_b=*/false);
  float s = cv[0] + cv[1] + cv[2] + cv[3] + cv[4] + cv[5] + cv[6] + cv[7];
  s += __shfl_xor(s, 16, 32);   // combine the two half-wave sums
  return s;                     // all lanes hold the full sum
}

__global__ void yolo_zero(int* __restrict__ ws, int n, float* __restrict__ out) {
  int i = blockIdx.x * blockDim.x + threadIdx.x;
  if (i < n) ws[i] = 0;
  if (i == 0) out[0] = 0.0f;
}

__global__ void yolo_scatter(const float* __restrict__ boxes,
                             int* __restrict__ map, int fs,
                             float aw0, float ah0, float aw1, float ah1,
                             float aw2, float ah2) {
  int tid = blockIdx.x * blockDim.x + threadIdx.x;
  if (tid >= BATCH * NTGT) return;
  int b = tid / NTGT;
  int t = tid - b * NTGT;
  const float* bx = boxes + (size_t)tid * 4;
  float ffs = (float)fs;
  float tbx = bx[0] * ffs, tby = bx[1] * ffs;
  float tw  = bx[2] * ffs, th  = bx[3] * ffs;
  float area = tw * th;
  // argmax IoU over 3 anchors (first max wins, like jnp.argmax)
  float best_iou = -1.0f; int best = 0;
  float aws[3] = {aw0, aw1, aw2};
  float ahs[3] = {ah0, ah1, ah2};
#pragma unroll
  for (int a = 0; a < 3; ++a) {
    float inter = fminf(tw, aws[a]) * fminf(th, ahs[a]);
    float uni   = area + aws[a] * ahs[a] - inter;
    float iou   = inter / (uni + 1e-6f);
    if (iou > best_iou) { best_iou = iou; best = a; }
  }
  int gx = (int)tbx; gx = min(max(gx, 0), fs - 1);
  int gy = (int)tby; gy = min(max(gy, 0), fs - 1);
  int flat = ((b * fs + gy) * fs + gx) * 3 + best;
  map[flat] = t + 1;  // racy last-writer-wins, mirrors jnp scatter .set
}

__global__ void yolo_main(const float* __restrict__ pred,
                          const int* __restrict__ map,
                          const float* __restrict__ boxes,
                          const int* __restrict__ cls,
                          float* __restrict__ out,
                          int fs, int N,
                          float aw0, float ah0, float aw1, float ah1,
                          float aw2, float ah2, float inv3N) {
  int cell = blockIdx.x * blockDim.x + threadIdx.x;
  float val = 0.0f;
  if (cell < N) {
    const float* p = pred + (size_t)cell * 25;
    int a = cell % 3;
    int x = (cell / 3) % fs;
    int y = (cell / (3 * fs)) % fs;
    int b = cell / (3 * fs * fs);
    float conf = sigm(p[4]);
    int m = map[cell];
    if (m == 0) {
      // no-object cell: only the L_NOOBJ conf BCE term survives the mask
      val = 0.5f * (-safelog(1.0f - conf)) * inv3N;
    } else {
      float aw = (a == 0) ? aw0 : ((a == 1) ? aw1 : aw2);
      float ah = (a == 0) ? ah0 : ((a == 1) ? ah1 : ah2);
      int t = m - 1;
      const float* bx = boxes + ((size_t)b * NTGT + t) * 4;
      float ffs = (float)fs;
      float tbx = bx[0] * ffs, tby = bx[1] * ffs;
      float tw  = bx[2] * ffs, th  = bx[3] * ffs;
      float tcx = tbx - (float)x;           // cell (x,y) == clipped int(tb)
      float tcy = tby - (float)y;           // by construction of the scatter
      float tww = logf(tw / aw + 1e-6f);
      float thh = logf(th / ah + 1e-6f);
      float pcx = sigm(p[0]) + (float)x;    // reference adds grid coords to
      float pcy = sigm(p[1]) + (float)y;    // pred but not to target (kept)
      float pw  = expf(p[2]) * aw;
      float ph  = expf(p[3]) * ah;
      float dcx = pcx - tcx, dcy = pcy - tcy;
      float dw  = pw - tww,  dh  = ph - thh;
      float sse = dcx * dcx + dcy * dcy + dw * dw + dh * dh;
      float bce_obj = -safelog(conf);
      int ci = cls[b * NTGT + t];
      float bce_cls = 0.0f;
#pragma unroll
      for (int c = 0; c < NCLS; ++c) {
        float s = sigm(p[5 + c]);
        float q = (c == ci) ? s : (1.0f - s);
        bce_cls += -safelog(q);
      }
      // per-scale loss = [5*sse + bce_obj + 0.5*bce_noobj]/N + bce_cls/(N*C)
      val = (5.0f * sse + bce_obj + bce_cls * (1.0f / NCLS)) * inv3N;
    }
  }
  // No early returns above: EXEC is all-ones here (WMMA requirement).
  float s = wave_sum_wmma(val);
  if ((threadIdx.x & 31) == 0) atomicAdd(out, s);
}

extern "C" void kernel_launch(void* const* d_in, const int* in_sizes, int n_in,
                              void* d_out, int out_size, void* d_ws, size_t ws_size,
                              hipStream_t stream) {
  (void)in_sizes; (void)n_in; (void)out_size; (void)ws_size;
  const float* pred0 = (const float*)d_in[0];   // (64,52,52,75)
  const float* pred1 = (const float*)d_in[1];   // (64,26,26,75)
  const float* pred2 = (const float*)d_in[2];   // (64,13,13,75)
  const float* boxes = (const float*)d_in[3];   // (64,50,4)
  const int*   cls   = (const int*)d_in[4];     // (64,50)
  float* out = (float*)d_out;

  const int N0 = BATCH * 52 * 52 * 3;  // 519168 (multiple of 256)
  const int N1 = BATCH * 26 * 26 * 3;  // 129792 (multiple of 256)
  const int N2 = BATCH * 13 * 13 * 3;  // 32448
  int* map0 = (int*)d_ws;
  int* map1 = map0 + N0;
  int* map2 = map1 + N1;
  const int Ntot = N0 + N1 + N2;

  // anchors / stride per scale: MASKS = [[6,7,8],[3,4,5],[0,1,2]]
  const float a0w0 = 14.5f,    a0h0 = 11.25f;    // 116,90  / 8
  const float a0w1 = 19.5f,    a0h1 = 24.75f;    // 156,198 / 8
  const float a0w2 = 46.625f,  a0h2 = 40.75f;    // 373,326 / 8
  const float a1w0 = 1.875f,   a1h0 = 3.8125f;   // 30,61   / 16
  const float a1w1 = 3.875f,   a1h1 = 2.8125f;   // 62,45   / 16
  const float a1w2 = 3.6875f,  a1h2 = 7.4375f;   // 59,119  / 16
  const float a2w0 = 0.3125f,  a2h0 = 0.40625f;  // 10,13   / 32
  const float a2w1 = 0.5f,     a2h1 = 0.9375f;   // 16,30   / 32
  const float a2w2 = 1.03125f, a2h2 = 0.71875f;  // 33,23   / 32

  yolo_zero<<<(Ntot + 255) / 256, 256, 0, stream>>>(map0, Ntot, out);

  const int BT = BATCH * NTGT;
  yolo_scatter<<<(BT + 255) / 256, 256, 0, stream>>>(
      boxes, map0, 52, a0w0, a0h0, a0w1, a0h1, a0w2, a0h2);
  yolo_scatter<<<(BT + 255) / 256, 256, 0, stream>>>(
      boxes, map1, 26, a1w0, a1h0, a1w1, a1h1, a1w2, a1h2);
  yolo_scatter<<<(BT + 255) / 256, 256, 0, stream>>>(
      boxes, map2, 13, a2w0, a2h0, a2w1, a2h1, a2w2, a2h2);

  yolo_main<<<(N0 + 255) / 256, 256, 0, stream>>>(
      pred0, map0, boxes, cls, out, 52, N0,
      a0w0, a0h0, a0w1, a0h1, a0w2, a0h2, 1.0f / (3.0f * (float)N0));
  yolo_main<<<(N1 + 255) / 256, 256, 0, stream>>>(
      pred1, map1, boxes, cls, out, 26, N1,
      a1w0, a1h0, a1w1, a1h1, a1w2, a1h2, 1.0f / (3.0f * (float)N1));
  yolo_main<<<(N2 + 255) / 256, 256, 0, stream>>>(
      pred2, map2, boxes, cls, out, 13, N2,
      a2w0, a2h0, a2w1, a2h1, a2w2, a2h2, 1.0f / (3.0f * (float)N2));
}